// BaseNetwork_42752104464634
// MI455X (gfx1250) — compile-verified
//
#include <hip/hip_runtime.h>
#include <stdint.h>
#include <math.h>

// Two-hot categorical projection, MI455X (gfx1250).
// Roofline: 630 MB of stores (4096*64*601 f32), ~1 MB of loads, trivial VALU.
// => pure store-bandwidth bound (~27 us @ 23.3 TB/s). No GEMM structure, so
// WMMA is inapplicable; the CDNA5 path used is the async LDS->global store
// engine (ASYNCcnt) with non-temporal, device-scope policy for the zero-fill.

#define NSUP            601
#define ROW_FLOATS      601
#define ROWS_PER_BLOCK  16
#define THREADS         256
#define CHUNK_FLOATS    (ROWS_PER_BLOCK * ROW_FLOATS)   // 9616 floats
#define CHUNK_VEC4      (CHUNK_FLOATS / 4)              // 2404 16B slots (chunk is 16B aligned)
#define EPSILON_F       0.001f

__global__ __launch_bounds__(THREADS)
void twohot_kernel(const float* __restrict__ target,
                   const float* __restrict__ supports,
                   float* __restrict__ out)
{
    // 4 KB LDS staging tile of zeros: source for the async zero-fill stores.
    __shared__ alignas(16) float zlds[THREADS * 4];

    const int tid = threadIdx.x;
    zlds[4 * tid + 0] = 0.0f;
    zlds[4 * tid + 1] = 0.0f;
    zlds[4 * tid + 2] = 0.0f;
    zlds[4 * tid + 3] = 0.0f;
    __syncthreads();   // DScnt drained: LDS zeros visible to the async engine

    // ---- Phase 1: zero-fill this block's chunk via async LDS->global stores ----
    const uint64_t blockBase = (uint64_t)blockIdx.x * (uint64_t)CHUNK_FLOATS;
    const uint64_t gbase     = (uint64_t)(uintptr_t)(out + blockBase);   // uniform -> SGPR pair
    const uint32_t lds_off   = (uint32_t)(uintptr_t)(&zlds[4 * tid]);    // per-lane LDS byte addr

    for (int slot = tid; slot < CHUNK_VEC4; slot += THREADS) {
        uint32_t voff = (uint32_t)slot * 16u;   // per-lane byte offset into the chunk
        // GVS mode: saddr = 64-bit base (SGPR pair), vaddr = 32-bit byte offset.
        // 16 bytes per lane per op, no VGPR data operand, tracked on ASYNCcnt.
        asm volatile(
            "global_store_async_from_lds_b128 %0, %1, %2 th:TH_STORE_NT scope:SCOPE_DEV"
            :
            : "v"(voff), "v"(lds_off), "s"(gbase)
            : "memory");
    }

    // ---- Phase 2a: two-hot computation, overlapped with the async drain ----
    // Everything except the final two stores is independent of the zero-fill,
    // so do it BEFORE waiting: the target/supports loads and the VALU work
    // hide under the outstanding async stores.
    float    p_low = 0.0f, p_high = 0.0f;
    float*   rp    = nullptr;
    int      li    = 0, ui = 0;
    const bool active = (tid < ROWS_PER_BLOCK);
    if (active) {
        const uint64_t row = (uint64_t)blockIdx.x * ROWS_PER_BLOCK + (uint64_t)tid;

        const float x   = target[row];
        const float sgn = (x > 0.0f) ? 1.0f : ((x < 0.0f) ? -1.0f : 0.0f);
        const float t   = sgn * (sqrtf(fabsf(x) + 1.0f) - 1.0f + EPSILON_F * x);

        // searchsorted(supports, t, side="right") - 1 on a uniform grid:
        // initial guess floor(t - s0), then +/-1 fix-up against the actual
        // array so bin edges match exactly under fp32 rounding.
        const float s0 = supports[0];
        li = (int)floorf(t - s0);
        li = (li < 0) ? 0 : ((li > NSUP - 1) ? NSUP - 1 : li);
        if (li < NSUP - 1 && supports[li + 1] <= t) {
            li += 1;
        } else if (li > 0 && supports[li] > t) {
            li -= 1;
        }
        ui = li + 1;
        if (ui > NSUP - 1) ui = NSUP - 1;

        const float lo = supports[li];
        const float hi = supports[ui];
        p_low  = (hi - t) / (hi - lo);
        p_high = 1.0f - p_low;
        rp     = out + row * (uint64_t)ROW_FLOATS;
    }

    // Wait for this wave's async stores to complete (device-scope ack), then
    // join all 8 waves: the whole chunk is now zero in global memory.
#if __has_builtin(__builtin_amdgcn_s_wait_asynccnt)
    __builtin_amdgcn_s_wait_asynccnt(0);
#else
    asm volatile("s_wait_asynccnt 0x0" ::: "memory");
#endif
    __syncthreads();

    // ---- Phase 2b: the two scatter stores ----
    if (active) {
        rp[li] = p_low;
        rp[ui] = p_high;   // same lane, later in program order: overwrites on collision
    }
}

extern "C" void kernel_launch(void* const* d_in, const int* in_sizes, int n_in,
                              void* d_out, int out_size, void* d_ws, size_t ws_size,
                              hipStream_t stream)
{
    const float* target   = (const float*)d_in[0];   // (4096, 64) float32
    const float* supports = (const float*)d_in[1];   // (601,)    float32
    float*       out      = (float*)d_out;           // (4096, 64, 601) float32

    const int rows = in_sizes[0];                    // 262144
    const int grid = rows / ROWS_PER_BLOCK;          // 16384 blocks

    hipLaunchKernelGGL(twohot_kernel, dim3(grid), dim3(THREADS), 0, stream,
                       target, supports, out);
}